// RecurrentGeneralizedPFAttention_85246510891772
// MI455X (gfx1250) — compile-verified
//
#include <hip/hip_runtime.h>
#include <hip/hip_bf16.h>

// MI455X (gfx1250) implementation, wave32 + V_WMMA_F32_16X16X4_F32 (full fp32,
// matching reference numerics; the workload is HBM-bound at 23.3 TB/s since the
// 134 MB attention tensor dominates, so fp32 matrix ops cost nothing vs bf16).

typedef __attribute__((ext_vector_type(2))) float v2f;
typedef __attribute__((ext_vector_type(8))) float v8f;

#define Bn   4
#define Nn   1024
#define En   512
#define Hn   8
#define Dn   64
#define HD   512          // H*D
#define QKV3 1536         // 3*H*D
#define SCALE_ 0.125f     // D^-0.5
#define THIRD_ (1.0f/3.0f)

__device__ __forceinline__ v8f wmma_f32(v2f a, v2f b, v8f c) {
  // D = A(16x4 f32) * B(4x16 f32) + C(16x16 f32)
  return __builtin_amdgcn_wmma_f32_16x16x4_f32(false, a, false, b,
                                               (short)0, c, false, false);
}

// ---------------------------------------------------------------------------
// Generic row-major GEMM: C[M,Nout] = A[M,K] @ W[K,Nout] (+ bias[Nout]).
// One wave computes a 16x64 strip (4 accumulators share the A fragment).
// Grid must supply exactly (M/16)*(Nout/64)/8 blocks of 256 threads.
// ---------------------------------------------------------------------------
__global__ __launch_bounds__(256) void gemm16_wmma(
    const float* __restrict__ A, const float* __restrict__ W,
    const float* __restrict__ bias, float* __restrict__ C,
    int M, int K, int Nout) {
  const int lane = threadIdx.x & 31;
  const int wave = threadIdx.x >> 5;
  const int rl   = lane & 15;     // row (A) or column (B/C) within tile
  const int hi   = lane >> 4;     // 0 or 1 (lane half)
  const int kh   = hi * 2;        // k sub-offset: lanes 0-15 -> k0/k1, 16-31 -> k2/k3
  const int strips_n = Nout >> 6;
  const int gw = blockIdx.x * 8 + wave;
  const int m0 = (gw / strips_n) << 4;
  const int n0 = (gw % strips_n) << 6;

  v8f acc0 = {}, acc1 = {}, acc2 = {}, acc3 = {};
  const float* Arow = A + (size_t)(m0 + rl) * K;
  for (int k0 = 0; k0 < K; k0 += 4) {
    v2f a = *(const v2f*)(Arow + k0 + kh);
    const float* W0 = W + (size_t)(k0 + kh) * Nout + n0 + rl;
    const float* W1 = W0 + Nout;
    v2f b0 = {W0[0],  W1[0]};
    v2f b1 = {W0[16], W1[16]};
    v2f b2 = {W0[32], W1[32]};
    v2f b3 = {W0[48], W1[48]};
    acc0 = wmma_f32(a, b0, acc0);
    acc1 = wmma_f32(a, b1, acc1);
    acc2 = wmma_f32(a, b2, acc2);
    acc3 = wmma_f32(a, b3, acc3);
  }

  float bs0 = 0.f, bs1 = 0.f, bs2 = 0.f, bs3 = 0.f;
  if (bias) {
    bs0 = bias[n0 + rl];      bs1 = bias[n0 + 16 + rl];
    bs2 = bias[n0 + 32 + rl]; bs3 = bias[n0 + 48 + rl];
  }
  #pragma unroll
  for (int v = 0; v < 8; ++v) {
    const int row = m0 + v + (hi << 3);   // C layout: VGPR v -> M=v (lanes<16) / v+8
    float* Crow = C + (size_t)row * Nout + n0 + rl;
    Crow[0]  = acc0[v] + bs0;
    Crow[16] = acc1[v] + bs1;
    Crow[32] = acc2[v] + bs2;
    Crow[48] = acc3[v] + bs3;
  }
}

// ---------------------------------------------------------------------------
// Fused score kernel: per (b,h,16-row band):
//   phase 1: S_sum = Q*(Rt^T) + Q*(Rc^T) + Q*(Rp^T) rows [i0,i0+16] -> LDS
//            (rows i0..i0+15 via WMMA, extra row i0+16 via VALU dots)
//   phase 2: AC tile via WMMA, combine with rel-shifted LDS reads, write
//            pre-softmax dots straight into d_out's [b,h,q,k] region.
// rel_shift(X)[i,j] = X[i, N-1+j-i]   (j<=i)
//                   = 0               (j==i+1)
//                   = X[i+1, j-i-2]   (j>=i+2)
// Grid: B*H*(N/16) blocks of 256 threads (8 waves).
// ---------------------------------------------------------------------------
__global__ __launch_bounds__(256) void attn_scores_kernel(
    const float* __restrict__ qkv,                      // [B*N, 1536]
    const float* __restrict__ rt, const float* __restrict__ rc,
    const float* __restrict__ rp,                       // [B*N, 512] each
    const float* __restrict__ bias_pf,                  // [H, D]
    float* __restrict__ att) {                          // [B,H,N,N] pre-softmax
  __shared__ float S[17][Nn];                           // 68 KB of 320 KB WGP LDS

  const int blk   = blockIdx.x;
  const int iband = blk & 63;            // N/16 = 64
  const int bh    = blk >> 6;
  const int h     = bh & (Hn - 1);
  const int b     = bh >> 3;
  const int i0    = iband << 4;

  const int lane = threadIdx.x & 31;
  const int wave = threadIdx.x >> 5;
  const int rl   = lane & 15;
  const int hi   = lane >> 4;
  const int kh   = hi * 2;

  const float* bb = bias_pf + h * Dn;

  // Preload Q band fragments (row i0+rl, all 16 k-steps), bias folded in.
  const float* qrow = qkv + (size_t)(b * Nn + i0 + rl) * QKV3 + h * Dn;
  v2f afrag[16];
  #pragma unroll
  for (int k = 0; k < 16; ++k) {
    const int dk = k * 4 + kh;
    v2f q = *(const v2f*)(qrow + dk);
    afrag[k].x = q.x + bb[dk];
    afrag[k].y = q.y + bb[dk + 1];
  }

  // ---- phase 1: S_sum rows 0..15 (WMMA; three R matrices into one acc) ----
  for (int job = wave; job < 64; job += 8) {
    const int j0 = job << 4;
    const size_t roff = (size_t)(b * Nn + j0 + rl) * HD + h * Dn;
    const float* Rt = rt + roff;
    const float* Rc = rc + roff;
    const float* Rp = rp + roff;
    v8f acc = {};
    #pragma unroll
    for (int k = 0; k < 16; ++k)
      acc = wmma_f32(afrag[k], *(const v2f*)(Rt + k * 4 + kh), acc);
    #pragma unroll
    for (int k = 0; k < 16; ++k)
      acc = wmma_f32(afrag[k], *(const v2f*)(Rc + k * 4 + kh), acc);
    #pragma unroll
    for (int k = 0; k < 16; ++k)
      acc = wmma_f32(afrag[k], *(const v2f*)(Rp + k * 4 + kh), acc);
    #pragma unroll
    for (int v = 0; v < 8; ++v)
      S[v + (hi << 3)][j0 + rl] = acc[v];
  }

  // ---- phase 1b: extra row i0+16 via plain dot products (4 cols/thread) ----
  {
    int r16 = i0 + 16; if (r16 > Nn - 1) r16 = Nn - 1;   // clamped row never read
    const float* q16 = qkv + (size_t)(b * Nn + r16) * QKV3 + h * Dn;
    for (int c = threadIdx.x; c < Nn; c += 256) {
      const size_t roff = (size_t)(b * Nn + c) * HD + h * Dn;
      const float* Rt = rt + roff;
      const float* Rc = rc + roff;
      const float* Rp = rp + roff;
      float s = 0.f;
      #pragma unroll 8
      for (int d = 0; d < Dn; ++d)
        s += (q16[d] + bb[d]) * (Rt[d] + Rc[d] + Rp[d]);
      S[16][c] = s;
    }
  }
  __syncthreads();

  // ---- phase 2: AC tiles + rel-shift combine, write dots ----
  float* outb = att + ((size_t)(b * Hn + h) * Nn + i0) * Nn;
  for (int job = wave; job < 64; job += 8) {
    const int j0 = job << 4;
    const float* Krow = qkv + (size_t)(b * Nn + j0 + rl) * QKV3 + HD + h * Dn;
    v8f acc = {};
    #pragma unroll
    for (int k = 0; k < 16; ++k)
      acc = wmma_f32(afrag[k], *(const v2f*)(Krow + k * 4 + kh), acc);
    #pragma unroll
    for (int v = 0; v < 8; ++v) {
      const int il = v + (hi << 3);        // local row 0..15
      const int i  = i0 + il;
      const int j  = j0 + rl;
      const int t  = j - i;
      float s;
      if (t <= 0)       s = S[il][Nn - 1 + t];
      else if (t == 1)  s = 0.f;
      else              s = S[il + 1][t - 2];
      outb[(size_t)il * Nn + j] = (acc[v] + s * THIRD_) * SCALE_;
    }
  }
}

// ---------------------------------------------------------------------------
// Softmax over the HEAD axis (faithful to the reference's axis=-1 on
// [q,k,b,h]).  att layout is [b,h,q,k]; each thread handles one (b,q,k) and
// reads H=8 values strided by N*N.
// ---------------------------------------------------------------------------
__global__ __launch_bounds__(256) void softmax_heads_kernel(float* __restrict__ att) {
  const size_t NN  = (size_t)Nn * Nn;
  const size_t idx = (size_t)blockIdx.x * 256 + threadIdx.x;   // over B*N*N
  const size_t b   = idx / NN;
  const size_t r   = idx - b * NN;
  float* p = att + b * (Hn * NN) + r;

  float vals[Hn];
  float mx = -3.4e38f;
  #pragma unroll
  for (int h = 0; h < Hn; ++h) { vals[h] = p[h * NN]; mx = fmaxf(mx, vals[h]); }
  float sum = 0.f;
  #pragma unroll
  for (int h = 0; h < Hn; ++h) { vals[h] = __expf(vals[h] - mx); sum += vals[h]; }
  const float inv = 1.0f / sum;
  #pragma unroll
  for (int h = 0; h < Hn; ++h) p[h * NN] = vals[h] * inv;
}

// ---------------------------------------------------------------------------
// out_pre[(b,i),(h,d)] = sum_j attn[b,h,i,j] * V[b,j,h,d]  (V = qkv cols 1024+)
// Grid: B*H*(N/16) blocks of 128 threads; wave w owns d-subtile w*16.
// ---------------------------------------------------------------------------
__global__ __launch_bounds__(128) void attnv_kernel(
    const float* __restrict__ att, const float* __restrict__ qkv,
    float* __restrict__ outp) {                         // [B*N, 512]
  const int blk   = blockIdx.x;
  const int iband = blk & 63;
  const int bh    = blk >> 6;
  const int h     = bh & (Hn - 1);
  const int b     = bh >> 3;
  const int i0    = iband << 4;

  const int lane = threadIdx.x & 31;
  const int wave = threadIdx.x >> 5;
  const int rl   = lane & 15;
  const int hi   = lane >> 4;
  const int kh   = hi * 2;
  const int d0   = wave << 4;

  const float* arow  = att + ((size_t)(b * Hn + h) * Nn + i0 + rl) * Nn;
  const float* vbase = qkv + (size_t)b * Nn * QKV3 + 2 * HD + h * Dn + d0 + rl;

  v8f acc = {};
  for (int j = 0; j < Nn; j += 4) {
    v2f a = *(const v2f*)(arow + j + kh);
    const float* vr = vbase + (size_t)(j + kh) * QKV3;
    v2f bf = {vr[0], vr[QKV3]};
    acc = wmma_f32(a, bf, acc);
  }
  #pragma unroll
  for (int v = 0; v < 8; ++v) {
    const int i = i0 + v + (hi << 3);
    outp[(size_t)(b * Nn + i) * HD + h * Dn + d0 + rl] = acc[v];
  }
}

// ---------------------------------------------------------------------------
// Host-side orchestration.
// Inputs: 0:x 1:r_t 2:r_c 3:r_p 4:bias_pf 5:w_qkv 6:w_krt 7:w_krc 8:w_krp
//         9:w_out 10:b_out 11:qlen
// d_out: out [4,1024,512] then attention [4,8,1024,1024]  (fp32)
// Workspace (floats): qkv 6291456 | rt 2097152 | rc | rp | out_pre 2097152
//   => 14,680,064 floats = 58.7 MB required in d_ws.
// ---------------------------------------------------------------------------
extern "C" void kernel_launch(void* const* d_in, const int* in_sizes, int n_in,
                              void* d_out, int out_size, void* d_ws, size_t ws_size,
                              hipStream_t stream) {
  const float* x       = (const float*)d_in[0];
  const float* r_t     = (const float*)d_in[1];
  const float* r_c     = (const float*)d_in[2];
  const float* r_p     = (const float*)d_in[3];
  const float* bias_pf = (const float*)d_in[4];
  const float* w_qkv   = (const float*)d_in[5];
  const float* w_krt   = (const float*)d_in[6];
  const float* w_krc   = (const float*)d_in[7];
  const float* w_krp   = (const float*)d_in[8];
  const float* w_out   = (const float*)d_in[9];
  const float* b_out   = (const float*)d_in[10];

  float* ws      = (float*)d_ws;
  float* ws_qkv  = ws;                       // 4096*1536
  float* ws_rt   = ws + 6291456;             // 4096*512
  float* ws_rc   = ws + 8388608;
  float* ws_rp   = ws + 10485760;
  float* ws_outp = ws + 12582912;            // 4096*512

  float* out_proj = (float*)d_out;                    // [4,1024,512]
  float* att      = (float*)d_out + (size_t)Bn * Nn * En;  // [4,8,1024,1024]

  const int M = Bn * Nn;   // 4096

  // 1) projections
  gemm16_wmma<<<(M / 16) * (QKV3 / 64) / 8, 256, 0, stream>>>(x,   w_qkv, nullptr, ws_qkv, M, En, QKV3);
  gemm16_wmma<<<(M / 16) * (HD   / 64) / 8, 256, 0, stream>>>(r_t, w_krt, nullptr, ws_rt,  M, En, HD);
  gemm16_wmma<<<(M / 16) * (HD   / 64) / 8, 256, 0, stream>>>(r_c, w_krc, nullptr, ws_rc,  M, En, HD);
  gemm16_wmma<<<(M / 16) * (HD   / 64) / 8, 256, 0, stream>>>(r_p, w_krp, nullptr, ws_rp,  M, En, HD);

  // 2) fused AC + rel-shifted BD scores -> pre-softmax dots in d_out
  attn_scores_kernel<<<Bn * Hn * (Nn / 16), 256, 0, stream>>>(ws_qkv, ws_rt, ws_rc, ws_rp, bias_pf, att);

  // 3) softmax over heads, in place
  softmax_heads_kernel<<<(Bn * Nn * Nn) / 256, 256, 0, stream>>>(att);

  // 4) attn @ V
  attnv_kernel<<<Bn * Hn * (Nn / 16), 128, 0, stream>>>(att, ws_qkv, ws_outp);

  // 5) output projection + bias -> first region of d_out
  gemm16_wmma<<<(M / 16) * (HD / 64) / 8, 256, 0, stream>>>(ws_outp, w_out, b_out, out_proj, M, HD, HD);
}